// L1Attn_11656541241666
// MI455X (gfx1250) — compile-verified
//
#include <hip/hip_runtime.h>
#include <stdint.h>

// Problem shape (fixed by the reference): q,k [2, 2048, 8, 64] f32,
// out [2, 2048, 2048, 8] f32 with out[b,i,j,h] = -1/8 * sum_w |q[b,j,h,w]-k[b,i,h,w]|
#define N_CTX    2048
#define NH       8
#define WD       64
#define QSTRIDE  (NH * WD)        // 512 floats between consecutive ctx rows
#define TILE     64               // i/j tile edge per workgroup
#define NTILES   (N_CTX / TILE)   // 32
#define ROWF     64               // floats per LDS row (no pad; swizzled)
#define OUT_ROW  (N_CTX * NH)     // 16384 floats per i row

typedef __attribute__((ext_vector_type(4))) float f32x4;

// Issue 4 async b128 copies (global -> LDS) per thread: one 64-row x 64-float tile
// per 256-thread block. LDS destination rows are chunk-swizzled:
//   physical_chunk = (logical_chunk + row) & 15
__device__ __forceinline__ void async_tile_load(const float* __restrict__ g0,
                                                unsigned lbase, int tid) {
  const int row = tid >> 2;                       // 0..63  (local ctx row)
  const int c0  = tid & 3;                        // 0..3
  const float* g = g0 + (size_t)row * QSTRIDE;    // start of this row's w-vector
#pragma unroll
  for (int u = 0; u < 4; ++u) {
    const int cs = c0 + 4 * u;                    // logical 16B chunk 0..15
    const int pc = (cs + row) & 15;               // swizzled physical chunk
    const unsigned l = lbase + (unsigned)(row * (ROWF * 4) + pc * 16);
    asm volatile("global_load_async_to_lds_b128 %0, %1, off"
                 :: "v"(l),
                    "v"((unsigned long long)(uintptr_t)(g + cs * 4))
                 : "memory");
  }
}

__global__ __launch_bounds__(256) void l1attn_kernel(const float* __restrict__ q,
                                                     const float* __restrict__ k,
                                                     float* __restrict__ out) {
  // [buf][Q/K][row][64 floats] : 2*2*64*64*4 = 64 KB LDS
  __shared__ float smem[2][2][TILE][ROWF];

  const int tid = threadIdx.x;
  const int tx  = tid & 15;        // j-lane   (16 lanes)
  const int ty  = tid >> 4;        // i-lane   (16 rows of threads)

  int blk = blockIdx.x;
  const int b  = blk / (NTILES * NTILES);
  blk -= b * (NTILES * NTILES);
  const int it = blk >> 5;         // i tile
  const int jt = blk & 31;         // j tile
  const int i0 = it * TILE;
  const int j0 = jt * TILE;

  const float* qb = q + (size_t)b * N_CTX * QSTRIDE + (size_t)j0 * QSTRIDE;
  const float* kb = k + (size_t)b * N_CTX * QSTRIDE + (size_t)i0 * QSTRIDE;
  float* ob = out + ((size_t)b * N_CTX + (size_t)i0) * OUT_ROW;

  int buf = 0;
  // Preload head 0 tiles (Q then K) via async tensor path.
  async_tile_load(qb, (unsigned)(uintptr_t)&smem[0][0][0][0], tid);
  async_tile_load(kb, (unsigned)(uintptr_t)&smem[0][1][0][0], tid);

  for (int h = 0; h < NH; ++h) {
    if (h + 1 < NH) {
      // Prefetch next head into the other buffer, then wait until only those
      // 8 just-issued copies remain outstanding (async loads complete in order).
      async_tile_load(qb + (h + 1) * WD,
                      (unsigned)(uintptr_t)&smem[buf ^ 1][0][0][0], tid);
      async_tile_load(kb + (h + 1) * WD,
                      (unsigned)(uintptr_t)&smem[buf ^ 1][1][0][0], tid);
      asm volatile("s_wait_asynccnt 8" ::: "memory");
    } else {
      asm volatile("s_wait_asynccnt 0" ::: "memory");
    }
    __syncthreads();

    const float* Qs = &smem[buf][0][0][0];
    const float* Ks = &smem[buf][1][0][0];

    float acc[4][4];
#pragma unroll
    for (int r = 0; r < 4; ++r)
#pragma unroll
      for (int s = 0; s < 4; ++s) acc[r][s] = 0.0f;

#pragma unroll 8
    for (int wc = 0; wc < 16; ++wc) {      // 4-wide w chunks
      // Swizzled physical chunk; (wc + row) & 15 with row = base + 16*r keeps
      // the same value for all r (16r & 15 == 0), so compute it once.
      const int pq = (wc + tx) & 15;
      const int pk = (wc + ty) & 15;
      f32x4 qf[4], kf[4];
#pragma unroll
      for (int r = 0; r < 4; ++r)          // rows tx, tx+16, tx+32, tx+48 -> 64 banks
        qf[r] = *(const f32x4*)(Qs + (tx + 16 * r) * ROWF + pq * 4);
#pragma unroll
      for (int s = 0; s < 4; ++s)          // rows ty+16s: uniform per half-wave (broadcast)
        kf[s] = *(const f32x4*)(Ks + (ty + 16 * s) * ROWF + pk * 4);
#pragma unroll
      for (int r = 0; r < 4; ++r)
#pragma unroll
        for (int s = 0; s < 4; ++s) {
          acc[r][s] += __builtin_fabsf(qf[r].x - kf[s].x);
          acc[r][s] += __builtin_fabsf(qf[r].y - kf[s].y);
          acc[r][s] += __builtin_fabsf(qf[r].z - kf[s].z);
          acc[r][s] += __builtin_fabsf(qf[r].w - kf[s].w);
        }
    }

    // out[b, i0+i, j0+j, h], h fastest (stride 1), j stride 8.
#pragma unroll
    for (int s = 0; s < 4; ++s) {
      const int i = ty + 16 * s;
      float* orow = ob + (size_t)i * OUT_ROW + (size_t)j0 * NH + h;
#pragma unroll
      for (int r = 0; r < 4; ++r) {
        const int j = tx + 16 * r;
        orow[(size_t)j * NH] = acc[r][s] * -0.125f;   // -1/sqrt(64)
      }
    }

    __syncthreads();   // all waves done reading buf before it is refilled
    buf ^= 1;
  }
}

extern "C" void kernel_launch(void* const* d_in, const int* in_sizes, int n_in,
                              void* d_out, int out_size, void* d_ws, size_t ws_size,
                              hipStream_t stream) {
  (void)in_sizes; (void)n_in; (void)out_size; (void)d_ws; (void)ws_size;
  const float* q = (const float*)d_in[0];
  const float* k = (const float*)d_in[1];
  float* out = (float*)d_out;
  const int nblocks = 2 * NTILES * NTILES;   // bs * (2048/64)^2 = 2048
  l1attn_kernel<<<nblocks, 256, 0, stream>>>(q, k, out);
}